// SheafODEFunc_71640054497902
// MI455X (gfx1250) — compile-verified
//
#include <hip/hip_runtime.h>
#include <hip/hip_bf16.h>
#include <math.h>

#define N_NODES 50000
#define N_EDGES 500000
#define DIM     32
#define EDIM    16
#define MSGIN   80          // 2*DIM + EDIM
#define MS      84          // padded LDS stride for msg-input tile / W1t (conflict-free, 8B aligned)
#define HS      36          // padded LDS stride for hidden tile / W2t
#define TILES_PER_BLOCK 4
#define EDGE_BLOCK 128      // 4 waves of 32

typedef __attribute__((ext_vector_type(2))) float v2f;
typedef __attribute__((ext_vector_type(8))) float v8f;

// ---------------------------------------------------------------------------
// Edge kernel: one wave32 handles a tile of 16 edges.
//   Phase 1: stage h[row], h[col], edge_attr into LDS
//   Phase 2: Qh matvec (lane = output row), streaming Q with b128 loads
//   Phase 3: MLP layer1 (K=80) + layer2 (K=32) via v_wmma_f32_16x16x4_f32
//   Phase 4: atomic scatter-add messages into agg[row]
// ---------------------------------------------------------------------------
__global__ __launch_bounds__(EDGE_BLOCK)
void sheaf_edge_kernel(const float* __restrict__ h,
                       const float* __restrict__ ea,
                       const float* __restrict__ Q,
                       const float* __restrict__ W1,
                       const float* __restrict__ b1,
                       const float* __restrict__ W2,
                       const float* __restrict__ b2,
                       const int*   __restrict__ eidx,
                       float*       __restrict__ agg)
{
    __shared__ float sW1t[DIM * MS];                    // W1 transposed: [n][k], k<80
    __shared__ float sW2t[DIM * HS];                    // W2 transposed: [n][k], k<32
    __shared__ float sMin [TILES_PER_BLOCK][16 * MS];   // msg_input tile [16 edges][80]
    __shared__ float sHcol[TILES_PER_BLOCK][16 * DIM];  // h[col] tile
    __shared__ float sHid [TILES_PER_BLOCK][16 * HS];   // hidden tile [16][32]

    const int tid = threadIdx.x;

    // Stage transposed weights once per block (shared by all waves)
    for (int i = tid; i < MSGIN * DIM; i += EDGE_BLOCK) {
        int k = i / DIM, n = i % DIM;
        sW1t[n * MS + k] = W1[i];
    }
    for (int i = tid; i < DIM * DIM; i += EDGE_BLOCK) {
        int k = i / DIM, n = i % DIM;
        sW2t[n * HS + k] = W2[i];
    }
    __syncthreads();

    const int wave = tid >> 5;
    const int lane = tid & 31;
    const int tile = blockIdx.x * TILES_PER_BLOCK + wave;
    if (tile >= N_EDGES / 16) return;

    const int eb = tile * 16;
    const int* __restrict__ rowp = eidx;
    const int* __restrict__ colp = eidx + N_EDGES;

    float* smin  = sMin[wave];
    float* shcol = sHcol[wave];
    float* shid  = sHid[wave];

    // ---- Phase 1: stage node states + edge attrs -------------------------
    for (int i = 0; i < 16; ++i) {
        const int e = eb + i;
        const int r = rowp[e];
        const int c = colp[e];
        smin [i * MS  + lane] = h[(size_t)r * DIM + lane];       // cols 0..31: h[row]
        shcol[i * DIM + lane] = h[(size_t)c * DIM + lane];
        if (lane < EDIM)
            smin[i * MS + 2 * DIM + lane] = ea[(size_t)e * EDIM + lane]; // cols 64..79
    }

    // ---- Phase 2: h_transported = Q[e] @ h[col]  (lane -> output row) ----
    for (int i = 0; i < 16; ++i) {
        const size_t qbase = ((size_t)(eb + i) * DIM + lane) * DIM;
        float acc = 0.f;
        #pragma unroll
        for (int j = 0; j < DIM; j += 4) {
            float4 q = *(const float4*)(Q + qbase + j);   // streams 2.05 GB total
            acc = fmaf(q.x, shcol[i * DIM + j    ], acc);
            acc = fmaf(q.y, shcol[i * DIM + j + 1], acc);
            acc = fmaf(q.z, shcol[i * DIM + j + 2], acc);
            acc = fmaf(q.w, shcol[i * DIM + j + 3], acc);
        }
        smin[i * MS + DIM + lane] = acc;                  // cols 32..63
    }

    // ---- Phase 3a: layer 1, M=16, K=80, N=32 (two 16-wide N tiles) -------
    const int mrow = lane & 15;           // A: M row / B: N col / D: N col
    const int koff = (lane >> 4) * 2;     // upper half-wave holds K+2,K+3
    v8f acc0 = {}; v8f acc1 = {};
    #pragma unroll
    for (int k = 0; k < MSGIN; k += 4) {
        v2f a   = *(const v2f*)(smin + mrow * MS + k + koff);
        v2f bn0 = *(const v2f*)(sW1t + (mrow     ) * MS + k + koff);
        v2f bn1 = *(const v2f*)(sW1t + (mrow + 16) * MS + k + koff);
        acc0 = __builtin_amdgcn_wmma_f32_16x16x4_f32(false, a, false, bn0, (short)0, acc0, false, false);
        acc1 = __builtin_amdgcn_wmma_f32_16x16x4_f32(false, a, false, bn1, (short)0, acc1, false, false);
    }

    // bias + ReLU, store hidden tile back to LDS (D layout -> row major)
    const int hi = (lane >> 4) * 8;       // upper half-wave owns rows 8..15
    const float bias0 = b1[mrow], bias1 = b1[mrow + 16];
    #pragma unroll
    for (int r = 0; r < 8; ++r) {
        float v0 = acc0[r] + bias0; v0 = v0 > 0.f ? v0 : 0.f;
        float v1 = acc1[r] + bias1; v1 = v1 > 0.f ? v1 : 0.f;
        shid[(r + hi) * HS + mrow     ] = v0;
        shid[(r + hi) * HS + mrow + 16] = v1;
    }

    // ---- Phase 3b: layer 2, M=16, K=32, N=32 -----------------------------
    v8f m0 = {}; v8f m1 = {};
    #pragma unroll
    for (int k = 0; k < DIM; k += 4) {
        v2f a   = *(const v2f*)(shid + mrow * HS + k + koff);
        v2f bn0 = *(const v2f*)(sW2t + (mrow     ) * HS + k + koff);
        v2f bn1 = *(const v2f*)(sW2t + (mrow + 16) * HS + k + koff);
        m0 = __builtin_amdgcn_wmma_f32_16x16x4_f32(false, a, false, bn0, (short)0, m0, false, false);
        m1 = __builtin_amdgcn_wmma_f32_16x16x4_f32(false, a, false, bn1, (short)0, m1, false, false);
    }

    // ---- Phase 4: segment-sum via L2-resident f32 atomics ----------------
    const float bb0 = b2[mrow], bb1 = b2[mrow + 16];
    #pragma unroll
    for (int r = 0; r < 8; ++r) {
        const int m   = r + hi;
        const int dst = rowp[eb + m];
        atomicAdd(agg + (size_t)dst * DIM + mrow,      m0[r] + bb0);
        atomicAdd(agg + (size_t)dst * DIM + mrow + 16, m1[r] + bb1);
    }
}

// ---------------------------------------------------------------------------
// Node kernel: tiny (0.3 GF, ~20 MB traffic) -> one thread per node, VALU.
// ---------------------------------------------------------------------------
__global__ __launch_bounds__(256)
void sheaf_node_kernel(const float* __restrict__ h,
                       const float* __restrict__ agg,
                       const float* __restrict__ Wd1,
                       const float* __restrict__ bd1,
                       const float* __restrict__ ln_g,
                       const float* __restrict__ ln_b,
                       const float* __restrict__ Wd2,
                       const float* __restrict__ bd2,
                       float*       __restrict__ out)
{
    __shared__ float sWd1[2 * DIM * DIM];
    __shared__ float sWd2[DIM * DIM];
    __shared__ float sB[4 * DIM];   // bd1 | ln_g | ln_b | bd2

    for (int i = threadIdx.x; i < 2 * DIM * DIM; i += 256) sWd1[i] = Wd1[i];
    for (int i = threadIdx.x; i < DIM * DIM; i += 256)     sWd2[i] = Wd2[i];
    if (threadIdx.x < DIM) {
        sB[threadIdx.x          ] = bd1 [threadIdx.x];
        sB[threadIdx.x + DIM    ] = ln_g[threadIdx.x];
        sB[threadIdx.x + 2 * DIM] = ln_b[threadIdx.x];
        sB[threadIdx.x + 3 * DIM] = bd2 [threadIdx.x];
    }
    __syncthreads();

    const int n = blockIdx.x * 256 + threadIdx.x;
    if (n >= N_NODES) return;

    float hv[DIM], av[DIM], x[DIM];
    #pragma unroll
    for (int k = 0; k < DIM; ++k) {
        hv[k] = h  [(size_t)n * DIM + k];
        av[k] = agg[(size_t)n * DIM + k];
    }

    float mu = 0.f;
    #pragma unroll
    for (int o = 0; o < DIM; ++o) {
        float acc = sB[o];
        #pragma unroll
        for (int k = 0; k < DIM; ++k) acc = fmaf(hv[k], sWd1[k * DIM + o], acc);
        #pragma unroll
        for (int k = 0; k < DIM; ++k) acc = fmaf(av[k], sWd1[(DIM + k) * DIM + o], acc);
        x[o] = acc;
        mu += acc;
    }
    mu *= (1.f / DIM);
    float var = 0.f;
    #pragma unroll
    for (int o = 0; o < DIM; ++o) { float d = x[o] - mu; var = fmaf(d, d, var); }
    var *= (1.f / DIM);
    const float rs = rsqrtf(var + 1e-5f);
    #pragma unroll
    for (int o = 0; o < DIM; ++o) {
        float v = (x[o] - mu) * rs * sB[DIM + o] + sB[2 * DIM + o];
        x[o] = v / (1.f + expf(-v));          // SiLU
    }
    #pragma unroll
    for (int o = 0; o < DIM; ++o) {
        float acc = sB[3 * DIM + o];
        #pragma unroll
        for (int k = 0; k < DIM; ++k) acc = fmaf(x[k], sWd2[k * DIM + o], acc);
        out[(size_t)n * DIM + o] = tanhf(acc);
    }
}

// ---------------------------------------------------------------------------
extern "C" void kernel_launch(void* const* d_in, const int* in_sizes, int n_in,
                              void* d_out, int out_size, void* d_ws, size_t ws_size,
                              hipStream_t stream)
{
    (void)in_sizes; (void)n_in; (void)out_size; (void)ws_size;
    // setup_inputs order: t,h,edge_attr,Q,W1,b1,W2,b2,Wd1,bd1,ln_g,ln_b,Wd2,bd2,edge_index
    const float* h   = (const float*)d_in[1];
    const float* ea  = (const float*)d_in[2];
    const float* Q   = (const float*)d_in[3];
    const float* W1  = (const float*)d_in[4];
    const float* b1  = (const float*)d_in[5];
    const float* W2  = (const float*)d_in[6];
    const float* b2  = (const float*)d_in[7];
    const float* Wd1 = (const float*)d_in[8];
    const float* bd1 = (const float*)d_in[9];
    const float* lng = (const float*)d_in[10];
    const float* lnb = (const float*)d_in[11];
    const float* Wd2 = (const float*)d_in[12];
    const float* bd2 = (const float*)d_in[13];
    const int*  eidx = (const int*)d_in[14];

    float* agg = (float*)d_ws;   // [N, D] accumulator, L2-resident (6.4 MB)
    hipMemsetAsync(agg, 0, (size_t)N_NODES * DIM * sizeof(float), stream);

    const int ntiles  = N_EDGES / 16;                                  // 31250
    const int nblocks = (ntiles + TILES_PER_BLOCK - 1) / TILES_PER_BLOCK;
    sheaf_edge_kernel<<<nblocks, EDGE_BLOCK, 0, stream>>>(h, ea, Q, W1, b1, W2, b2, eidx, agg);

    sheaf_node_kernel<<<(N_NODES + 255) / 256, 256, 0, stream>>>(
        h, agg, Wd1, bd1, lng, lnb, Wd2, bd2, (float*)d_out);
}